// GraphSAGE_39041252721036
// MI455X (gfx1250) — compile-verified
//
#include <hip/hip_runtime.h>
#include <cstdint>

// GraphSAGE on MI455X (gfx1250, wave32).
// Bottleneck is edge gather + f32 atomic scatter (~2.5 GB traffic @ 23.3 TB/s).
// GEMMs (5 GFLOP total) run on V_WMMA_F32_16X16X4_F32 to keep full f32 precision.

typedef __attribute__((ext_vector_type(2))) float v2f;
typedef __attribute__((ext_vector_type(8))) float v8f;

#define N_NODES 100000
#define N_EDGES 1600000
#define FDIM    64
#define FINAL   192   // H * 3 (JK concat)

// ---------------- zeroing (float4 grid) ----------------
__global__ void zero_f4(float* __restrict__ p, long n4) {
    long i = (long)blockIdx.x * blockDim.x + threadIdx.x;
    if (i < n4) ((float4*)p)[i] = make_float4(0.f, 0.f, 0.f, 0.f);
}

// ---------------- degree / inv_deg ----------------
__global__ void degree_kernel(const int* __restrict__ dst, float* __restrict__ deg) {
    int e = blockIdx.x * blockDim.x + threadIdx.x;
    if (e < N_EDGES) unsafeAtomicAdd(&deg[dst[e]], 1.0f);
}

__global__ void invdeg_kernel(float* __restrict__ deg) {
    int n = blockIdx.x * blockDim.x + threadIdx.x;
    if (n < N_NODES) deg[n] = 1.0f / fmaxf(deg[n], 1.0f);
}

// ---------------- edge scatter: agg[dst] += h[src] ----------------
// 16 threads per edge, each thread moves a float4 (coalesced 256B per edge).
__global__ void scatter_kernel(const float* __restrict__ h, int hstride,
                               const int* __restrict__ src, const int* __restrict__ dst,
                               float* __restrict__ agg) {
    long idx = (long)blockIdx.x * blockDim.x + threadIdx.x;
    if (idx >= (long)N_EDGES * 16) return;
    int e  = (int)(idx >> 4);
    int fb = (int)(idx & 15);
    int s = src[e], d = dst[e];
    const float4 v = *(const float4*)(h + (long)s * hstride + fb * 4);
    float* ap = agg + (long)d * FDIM + fb * 4;
    unsafeAtomicAdd(ap + 0, v.x);
    unsafeAtomicAdd(ap + 1, v.y);
    unsafeAtomicAdd(ap + 2, v.z);
    unsafeAtomicAdd(ap + 3, v.w);
}

// ---------------- fused SAGE layer GEMM ----------------
// h_out = relu( (agg * inv_deg) @ Wl + bias + h_prev @ Wr )
// One K=128 concat-GEMM: A = [agg*inv | h_prev], W = [Wl ; Wr] (staged in LDS).
// One wave handles 16 rows x 64 cols via 4 accumulator tiles of 16x16.
__global__ __launch_bounds__(256) void sage_gemm(
    const float* __restrict__ agg, const float* __restrict__ invdeg,
    const float* __restrict__ hprev, int hstride,
    const float* __restrict__ Wl, const float* __restrict__ Wr,
    const float* __restrict__ bias,
    float* __restrict__ hout /* = hcat + layer*64, row stride FINAL */) {

    __shared__ float lw[128 * 64];                 // [Wl ; Wr], 32 KB
    for (int i = threadIdx.x; i < 64 * 64; i += 256) {
        lw[i]        = Wl[i];
        lw[4096 + i] = Wr[i];
    }
    __syncthreads();

    const int lane  = threadIdx.x & 31;
    const int wave  = threadIdx.x >> 5;
    const int rit   = lane & 15;                   // row-in-tile (A) / col-in-tile (B)
    const int khalf = lane >> 4;                   // 0: K {k,k+1}; 1: K {k+2,k+3}
    const int m0    = blockIdx.x * 128 + wave * 16;
    const int row   = m0 + rit;
    const int rowc  = row < N_NODES ? row : N_NODES - 1;  // clamp: EXEC must stay all-1s
    const float inv = invdeg[rowc];

    v8f acc[4] = {};

    // K = 0..63 : mean-aggregated neighbors (scaled on the fly)
    const float2* arow = (const float2*)(agg + (long)rowc * FDIM);
    for (int kk = 0; kk < 64; kk += 4) {
        const int k0 = kk + khalf * 2;
        float2 t = arow[k0 >> 1];
        v2f a; a.x = t.x * inv; a.y = t.y * inv;
        #pragma unroll
        for (int j = 0; j < 4; ++j) {
            const int col = j * 16 + rit;
            v2f b; b.x = lw[k0 * 64 + col]; b.y = lw[(k0 + 1) * 64 + col];
            acc[j] = __builtin_amdgcn_wmma_f32_16x16x4_f32(
                false, a, false, b, (short)0, acc[j], false, false);
        }
    }
    // K = 64..127 : root features
    const float* hr = hprev + (long)rowc * hstride;
    for (int kk = 0; kk < 64; kk += 4) {
        const int k0 = kk + khalf * 2;
        float2 t = *(const float2*)(hr + k0);
        v2f a; a.x = t.x; a.y = t.y;
        #pragma unroll
        for (int j = 0; j < 4; ++j) {
            const int col = j * 16 + rit;
            v2f b; b.x = lw[(64 + k0) * 64 + col]; b.y = lw[(64 + k0 + 1) * 64 + col];
            acc[j] = __builtin_amdgcn_wmma_f32_16x16x4_f32(
                false, a, false, b, (short)0, acc[j], false, false);
        }
    }

    // Epilogue: bias + relu, scatter per C/D layout (VGPR v: M=v (+8 for hi lanes))
    #pragma unroll
    for (int j = 0; j < 4; ++j) {
        const int col = j * 16 + rit;
        const float bb = bias[col];
        #pragma unroll
        for (int v = 0; v < 8; ++v) {
            const int r = m0 + v + khalf * 8;
            if (r < N_NODES) {
                float val = acc[j][v] + bb;
                hout[(long)r * FINAL + col] = val > 0.f ? val : 0.f;
            }
        }
    }
}

// ---------------- JK-cat linear + sigmoid ----------------
// One wave per node: coalesced 192-float read, shfl_xor wave32 reduction.
__global__ void final_kernel(const float* __restrict__ hcat,
                             const float* __restrict__ Wlin,
                             const float* __restrict__ blin,
                             float* __restrict__ out) {
    int t = blockIdx.x * blockDim.x + threadIdx.x;
    int n = t >> 5;                 // wave-uniform
    int lane = t & 31;
    if (n >= N_NODES) return;
    const float* hr = hcat + (long)n * FINAL;
    float s = 0.f;
    #pragma unroll
    for (int i = 0; i < FINAL / 32; ++i) {
        int idx = lane + i * 32;
        s += hr[idx] * Wlin[idx];
    }
    #pragma unroll
    for (int off = 16; off > 0; off >>= 1) s += __shfl_xor(s, off, 32);
    if (lane == 0) {
        float z = s + blin[0];
        out[n] = 1.0f / (1.0f + expf(-z));
    }
}

extern "C" void kernel_launch(void* const* d_in, const int* in_sizes, int n_in,
                              void* d_out, int out_size, void* d_ws, size_t ws_size,
                              hipStream_t stream) {
    const float* x   = (const float*)d_in[0];
    const int*   ei  = (const int*)d_in[1];
    const int*   src = ei;
    const int*   dst = ei + N_EDGES;
    const float* Wl[3] = {(const float*)d_in[2], (const float*)d_in[5], (const float*)d_in[8]};
    const float* bs[3] = {(const float*)d_in[3], (const float*)d_in[6], (const float*)d_in[9]};
    const float* Wr[3] = {(const float*)d_in[4], (const float*)d_in[7], (const float*)d_in[10]};
    const float* Wlin = (const float*)d_in[11];
    const float* blin = (const float*)d_in[12];
    float* out = (float*)d_out;

    // workspace: inv_deg [N] | agg [N*64] | hcat [N*192]  (~103 MB)
    float* invdeg = (float*)d_ws;
    float* agg    = invdeg + N_NODES;
    float* hcat   = agg + (size_t)N_NODES * FDIM;

    // degree -> inv_deg
    zero_f4<<<(N_NODES / 4 + 255) / 256, 256, 0, stream>>>(invdeg, N_NODES / 4);
    degree_kernel<<<(N_EDGES + 255) / 256, 256, 0, stream>>>(dst, invdeg);
    invdeg_kernel<<<(N_NODES + 255) / 256, 256, 0, stream>>>(invdeg);

    const float* hprev = x;
    int hstride = FDIM;
    for (int l = 0; l < 3; ++l) {
        long aggn4 = (long)N_NODES * FDIM / 4;
        zero_f4<<<(int)((aggn4 + 255) / 256), 256, 0, stream>>>(agg, aggn4);
        scatter_kernel<<<(int)(((long)N_EDGES * 16 + 255) / 256), 256, 0, stream>>>(
            hprev, hstride, src, dst, agg);
        sage_gemm<<<(N_NODES + 127) / 128, 256, 0, stream>>>(
            agg, invdeg, hprev, hstride, Wl[l], Wr[l], bs[l], hcat + l * FDIM);
        hprev = hcat + l * FDIM;
        hstride = FINAL;
    }

    final_kernel<<<(int)(((long)N_NODES * 32 + 255) / 256), 256, 0, stream>>>(
        hcat, Wlin, blin, out);
}